// GraphAttentionLayer_25357486916381
// MI455X (gfx1250) — compile-verified
//
#include <hip/hip_runtime.h>
#include <hip/hip_bf16.h>
#include <math.h>

typedef float v2f __attribute__((ext_vector_type(2)));
typedef float v8f __attribute__((ext_vector_type(8)));

#define NEGV (-9000000000000000.0f)

// One wave per 16-row tile (WMMA M=16). 8 waves per block -> 128 rows of one batch.
__global__ __launch_bounds__(256, 1) void gat_wmma_kernel(
    const float* __restrict__ x,    // (bs, N)   [input, last dim 1 squeezed]
    const int*   __restrict__ adj,  // (N, N)
    const float* __restrict__ W,    // (F,)
    const float* __restrict__ a,    // (2F,)
    float*       __restrict__ out,  // (bs, N, F)
    int bs, int N, int F)
{
    __shared__ float xs[1024];      // x[b, :]
    __shared__ float Ws[128];       // W
    __shared__ float nums[8][16];
    __shared__ float dens[8][16];
    __shared__ float ts[128];       // t per row of this block

    const int tid  = threadIdx.x;
    const int wave = tid >> 5;
    const int lane = tid & 31;

    const int tilesPerBatch = N / 128;           // 8
    const int b       = blockIdx.x / tilesPerBatch;
    const int rowBase = (blockIdx.x % tilesPerBatch) * 128;
    const int i0      = rowBase + wave * 16;     // this wave's first row

    const float* xb = x + (size_t)b * N;
    for (int j = tid; j < N; j += 256) xs[j] = xb[j];
    for (int f = tid; f < F && f < 128; f += 256) Ws[f] = W[f];
    __syncthreads();

    // c1 = W . a[:F], c2 = W . a[F:]  (wave-redundant, tiny)
    float p1 = 0.f, p2 = 0.f;
    for (int f = lane; f < F; f += 32) {
        float w = Ws[f];
        p1 += w * a[f];
        p2 += w * a[F + f];
    }
    #pragma unroll
    for (int off = 16; off; off >>= 1) {
        p1 += __shfl_xor(p1, off, 32);
        p2 += __shfl_xor(p2, off, 32);
    }
    const float c1 = p1, c2 = p2;

    // WMMA 32-bit A(16x4) layout: lane L -> row r=L&15; half=L>>4;
    //   VGPR0 holds K=2*half, VGPR1 holds K=2*half+1.
    const int   r    = lane & 15;
    const int   half = lane >> 4;
    const int   row  = i0 + r;
    const float s1r  = c1 * xs[row];
    const int*  adjRow = adj + (size_t)row * N + 2 * half;

    // ---- pass 1: row max over kept entries (init NEG; matches jax softmax) ----
    float m = NEGV;
    for (int j0 = 0; j0 < N; j0 += 4) {
        int2  av = *(const int2*)(adjRow + j0);
        float xa = xs[j0 + 2 * half];
        float xc = xs[j0 + 2 * half + 1];
        float sa = fmaf(c2, xa, s1r);
        float sb = fmaf(c2, xc, s1r);
        m = fmaxf(m, (av.x > 0 && sa > 0.f) ? sa : NEGV);
        m = fmaxf(m, (av.y > 0 && sb > 0.f) ? sb : NEGV);
    }
    m = fmaxf(m, __shfl_xor(m, 16, 32));   // lanes L and L+16 share row r

    // ---- pass 2: WMMA-accumulate C = att_tile @ [x | 1] -> (num, den) ----
    // B(4x16) layout: lane L -> col n=L&15; VGPR0 holds K=2*half, VGPR1 K=2*half+1.
    const bool isN0 = (r == 0);
    const bool isN1 = (r == 1);
    v8f acc = {};
    for (int j0 = 0; j0 < N; j0 += 4) {
        int2  av = *(const int2*)(adjRow + j0);
        float xa = xs[j0 + 2 * half];
        float xc = xs[j0 + 2 * half + 1];
        float sa = fmaf(c2, xa, s1r);
        float sb = fmaf(c2, xc, s1r);
        // exp((kept ? s : NEG) - m): underflows to exactly 0 for dropped entries
        // when m>NEG; gives exp(0)=1 uniformly when the whole row is masked.
        float ea = __expf(((av.x > 0 && sa > 0.f) ? sa : NEGV) - m);
        float eb = __expf(((av.y > 0 && sb > 0.f) ? sb : NEGV) - m);
        v2f A; A.x = ea; A.y = eb;
        v2f B;
        B.x = isN0 ? xa : (isN1 ? 1.0f : 0.0f);
        B.y = isN0 ? xc : (isN1 ? 1.0f : 0.0f);
        acc = __builtin_amdgcn_wmma_f32_16x16x4_f32(
                  false, A, false, B, (short)0, acc, false, false);
    }

    // C(16x16) layout: VGPR v -> row M=v (lanes 0-15) / M=v+8 (lanes 16-31), col N=lane&15.
    // Column 0 = numerator, column 1 = denominator.
    if (r == 0) {
        #pragma unroll
        for (int v = 0; v < 8; ++v) nums[wave][half * 8 + v] = acc[v];
    }
    if (r == 1) {
        #pragma unroll
        for (int v = 0; v < 8; ++v) dens[wave][half * 8 + v] = acc[v];
    }
    __syncthreads();

    if (lane < 16) {
        // den >= exp(0) = 1 always (max element contributes 1; masked rows sum to N)
        ts[wave * 16 + lane] = nums[wave][lane] / dens[wave][lane];
    }
    __syncthreads();

    // ---- output: out[b, rowBase+rr, f] = elu(t[rr] * W[f]) ----
    float* outB = out + ((size_t)b * N + rowBase) * F;
    const int total = 128 * F;
    for (int idx = tid; idx < total; idx += 256) {
        int   rr = idx / F;
        int   f  = idx - rr * F;
        float v  = ts[rr] * Ws[f];
        outB[idx] = v > 0.f ? v : expm1f(v);
    }
}

extern "C" void kernel_launch(void* const* d_in, const int* in_sizes, int n_in,
                              void* d_out, int out_size, void* d_ws, size_t ws_size,
                              hipStream_t stream) {
    const float* x   = (const float*)d_in[0];  // input (bs,N,1)
    const int*   adj = (const int*)d_in[1];    // adj (N,N)
    // d_in[2] = ext_input, d_in[3] = side_input: unused by the reference
    const float* W   = (const float*)d_in[4];  // (1,F)
    const float* a   = (const float*)d_in[5];  // (2F,1)
    float* out = (float*)d_out;

    int F = in_sizes[4];                       // 128
    int N = 1;
    { long s = in_sizes[1]; while ((long)N * N < s) ++N; }   // N = 1024
    int bs = in_sizes[0] / N;                  // 32

    dim3 grid(bs * (N / 128));
    dim3 block(256);
    gat_wmma_kernel<<<grid, block, 0, stream>>>(x, adj, W, a, out, bs, N, F);
}